// DTMTIR_79559974191225
// MI455X (gfx1250) — compile-verified
//
#include <hip/hip_runtime.h>
#include <math.h>

// Problem constants (from reference)
#define KD 64
#define TD 100
#define VD 10000
constexpr int   V4      = VD / 4;                 // 2500 float4 per row
constexpr int   BLOCK   = 256;                    // 8 wave32 waves
constexpr int   NIT     = (V4 + BLOCK - 1) / BLOCK; // 10 float4 per thread
constexpr float DELTA   = 0.005f;
constexpr float EPS_DIV = 1e-6f;
constexpr float INV_D   = 1.0f / (DELTA + EPS_DIV);
constexpr float INV_1   = 1.0f / (1.0f + EPS_DIV);
constexpr float LOG_DELTA = -5.2983173665480366e+0f; // ln(0.005)

typedef __attribute__((ext_vector_type(2))) float v2f;
typedef __attribute__((ext_vector_type(8))) float v8f;

// Wave-wide (32 lane) additive reduction using V_WMMA_F32_16X16X4_F32.
// A (16x4 f32, 2 VGPRs): lanes0-15 of v0 -> A[m=lane,k=0], lanes16-31 of v0 -> A[m,k=2];
// v1 (k=1,3) set to zero. B = all ones (any layout, every element is 1).
// D[m,n] = sum_k A[m,k] = s[m] + s[m+16].
// C/D layout: vgpr j holds D[j, n=lane] for lanes0-15 and D[j+8, n] for lanes16-31, so
// sum over the 8 D vgprs gives  sum_{m=0..7}(s[m]+s[m+16])  in lanes 0-15 and
// sum_{m=8..15}(s[m]+s[m+16]) in lanes 16-31; one xor-16 shuffle finishes the reduction.
__device__ __forceinline__ float wave_sum_wmma(float x) {
  v2f a; a[0] = x;    a[1] = 0.0f;
  v2f b; b[0] = 1.0f; b[1] = 1.0f;
  v8f c = {};
  v8f d = __builtin_amdgcn_wmma_f32_16x16x4_f32(
      /*neg_a=*/false, a, /*neg_b=*/false, b,
      /*c_mod=*/(short)0, c, /*reuse_a=*/false, /*reuse_b=*/false);
  float r = d[0] + d[1] + d[2] + d[3] + d[4] + d[5] + d[6] + d[7];
  r += __shfl_xor(r, 16);
  return r; // all 32 lanes hold the wave total
}

// Fused: beta = mu + exp(0.5*ls)*eps ; per-row softmax over V ; per-block KL partial.
// One block per (t,k) row. Recomputes beta[t-1] (t-1 slice is L2-resident: 7.7MB << 192MB).
__global__ __launch_bounds__(BLOCK) void beta_softmax_kl_kernel(
    const float* __restrict__ mu, const float* __restrict__ ls,
    const float* __restrict__ eps, float* __restrict__ out,
    float* __restrict__ klpart)
{
  const int b    = blockIdx.x;      // b = t*K + k  -> consecutive t, good L2 temporal reuse
  const int t    = b / KD;
  const int k    = b - t * KD;
  const int tid  = threadIdx.x;
  const int lane = tid & 31;
  const int wid  = tid >> 5;

  const float4* mu4 = (const float4*)(mu  + ((size_t)k * TD + t) * VD);
  const float4* ls4 = (const float4*)(ls  + ((size_t)k * TD + t) * VD);
  const float4* ep4 = (const float4*)(eps + ((size_t)t * KD + k) * VD);
  float4*      out4 = (float4*)     (out + ((size_t)t * KD + k) * VD);

  float4 breg[NIT];
  float  lmax = -INFINITY;
  float  acc  = 0.0f;

  if (t == 0) {
#define KL0_COMP(c) do {                                                     \
      float sv = __expf(0.5f * l.c); float bv = fmaf(sv, e.c, m.c);          \
      bt.c = bv; lmax = fmaxf(lmax, bv);                                     \
      acc += fmaf(sv, sv, m.c * m.c) * INV_1 - 1.0f - l.c; } while (0)
#pragma unroll
    for (int i = 0; i < NIT; ++i) {
      int idx = i * BLOCK + tid;
      if (idx < V4) {
        float4 m = mu4[idx], l = ls4[idx], e = ep4[idx];
        float4 bt;
        KL0_COMP(x); KL0_COMP(y); KL0_COMP(z); KL0_COMP(w);
        breg[i] = bt;
      }
    }
  } else {
    const float4* mup4 = (const float4*)(mu  + ((size_t)k * TD + (t - 1)) * VD);
    const float4* lsp4 = (const float4*)(ls  + ((size_t)k * TD + (t - 1)) * VD);
    const float4* epp4 = (const float4*)(eps + ((size_t)(t - 1) * KD + k) * VD);
#define KL1_COMP(c) do {                                                     \
      float sv = __expf(0.5f * l.c); float bv = fmaf(sv, e.c, m.c);          \
      bt.c = bv; lmax = fmaxf(lmax, bv);                                     \
      float bp = fmaf(__expf(0.5f * lp.c), ee.c, mp.c);                      \
      float df = m.c - bp;                                                   \
      acc += fmaf(sv, sv, df * df) * INV_D + (LOG_DELTA - 1.0f) - l.c; } while (0)
#pragma unroll
    for (int i = 0; i < NIT; ++i) {
      int idx = i * BLOCK + tid;
      if (idx < V4) {
        float4 m  = mu4[idx],  l  = ls4[idx],  e  = ep4[idx];
        float4 mp = mup4[idx], lp = lsp4[idx], ee = epp4[idx];
        float4 bt;
        KL1_COMP(x); KL1_COMP(y); KL1_COMP(z); KL1_COMP(w);
        breg[i] = bt;
      }
    }
  }

  __shared__ float red[8];

  // ---- block max (shuffle + LDS) ----
#pragma unroll
  for (int off = 16; off > 0; off >>= 1)
    lmax = fmaxf(lmax, __shfl_xor(lmax, off));
  if (lane == 0) red[wid] = lmax;
  __syncthreads();
  float rmax = red[0];
#pragma unroll
  for (int i = 1; i < 8; ++i) rmax = fmaxf(rmax, red[i]);
  __syncthreads();

  // ---- exp + row sum (WMMA wave reduce + LDS) ----
  float lsum = 0.0f;
#pragma unroll
  for (int i = 0; i < NIT; ++i) {
    int idx = i * BLOCK + tid;
    if (idx < V4) {
      float4 bt = breg[i];
      bt.x = __expf(bt.x - rmax); bt.y = __expf(bt.y - rmax);
      bt.z = __expf(bt.z - rmax); bt.w = __expf(bt.w - rmax);
      lsum += (bt.x + bt.y) + (bt.z + bt.w);
      breg[i] = bt;
    }
  }
  float wsum = wave_sum_wmma(lsum);
  if (lane == 0) red[wid] = wsum;
  __syncthreads();
  float rsum = 0.0f;
#pragma unroll
  for (int i = 0; i < 8; ++i) rsum += red[i];
  const float rinv = 1.0f / rsum;
  __syncthreads();

  // ---- normalize + store ----
#pragma unroll
  for (int i = 0; i < NIT; ++i) {
    int idx = i * BLOCK + tid;
    if (idx < V4) {
      float4 bt = breg[i];
      bt.x *= rinv; bt.y *= rinv; bt.z *= rinv; bt.w *= rinv;
      out4[idx] = bt;
    }
  }

  // ---- KL partial for this block (WMMA wave reduce + LDS) ----
  float wk = wave_sum_wmma(acc);
  if (lane == 0) red[wid] = wk;
  __syncthreads();
  if (tid == 0) {
    float s = 0.0f;
#pragma unroll
    for (int i = 0; i < 8; ++i) s += red[i];
    klpart[b] = 0.5f * s;
  }
}

// Reduce the 6400 per-block KL partials into the final scalar at out[T*K*V].
__global__ __launch_bounds__(256) void kl_reduce_kernel(
    const float* __restrict__ part, float* __restrict__ out_kl, int n)
{
  const int tid  = threadIdx.x;
  const int lane = tid & 31;
  const int wid  = tid >> 5;
  float a = 0.0f;
  for (int i = tid; i < n; i += 256) a += part[i];
  float w = wave_sum_wmma(a);
  __shared__ float red[8];
  if (lane == 0) red[wid] = w;
  __syncthreads();
  if (tid == 0) {
    float s = 0.0f;
#pragma unroll
    for (int i = 0; i < 8; ++i) s += red[i];
    *out_kl = s;
  }
}

extern "C" void kernel_launch(void* const* d_in, const int* in_sizes, int n_in,
                              void* d_out, int out_size, void* d_ws, size_t ws_size,
                              hipStream_t stream) {
  (void)in_sizes; (void)n_in; (void)out_size; (void)ws_size;
  const float* mu = (const float*)d_in[0];   // mu_q_beta  [K,T,V]
  const float* ls = (const float*)d_in[1];   // logsig_q_beta [K,T,V]
  const float* ep = (const float*)d_in[2];   // eps [T,K,V]
  float* out    = (float*)d_out;             // softmax betas [T,K,V] ++ kl scalar
  float* klpart = (float*)d_ws;              // 6400 floats of scratch

  beta_softmax_kl_kernel<<<TD * KD, BLOCK, 0, stream>>>(mu, ls, ep, out, klpart);
  kl_reduce_kernel<<<1, 256, 0, stream>>>(klpart, out + (size_t)TD * KD * VD, TD * KD);
}